// DiscriptorRententionLoss_29214367547795
// MI455X (gfx1250) — compile-verified
//
#include <hip/hip_runtime.h>
#include <math.h>

// ---------------------------------------------------------------------------
// DescriptorRetentionLoss for MI455X (gfx1250), wave32 + WMMA f32 16x16x4.
//
// Inputs (setup_inputs order):
//  0 points [B,M,2] f32        4 descriptors [B,M,D] f32
//  1 depth  [B,1,H,W] f32      5 mem_pts [N,3] f32
//  2 pose   [B,3,4] f32        6 mem_desc [N,D] f32
//  3 K      [B,3,3] f32        7 next_ptr scalar i32
// Output (flat f32): [loss(1), n_match(1), new_mem_pts(N*3), new_mem_desc(N*D)]
// ---------------------------------------------------------------------------

#define EMA_F   0.5f
#define EPS_F   1e-8f
#define THR2_F  0.01f

typedef float v2f __attribute__((ext_vector_type(2)));
typedef float v8f __attribute__((ext_vector_type(8)));

__device__ __forceinline__ float wred_sum(float x) {
  #pragma unroll
  for (int o = 16; o > 0; o >>= 1) x += __shfl_xor(x, o, 32);
  return x;
}

// ---- 1) Stage A matrix: memA[i] = (-2x, -2y, -2z, |m|^2); pad rows -> +inf.
//         Also copies mem_pts into the output buffer.
__global__ void k_prep(const float* __restrict__ mem_pts, float4* __restrict__ memA,
                       float* __restrict__ out_pts, int N, int NpadAll) {
  int i = blockIdx.x * blockDim.x + threadIdx.x;
  if (i >= NpadAll) return;
  if (i < N) {
    float x = mem_pts[i * 3 + 0], y = mem_pts[i * 3 + 1], z = mem_pts[i * 3 + 2];
    memA[i] = make_float4(-2.0f * x, -2.0f * y, -2.0f * z, x * x + y * y + z * z);
    out_pts[i * 3 + 0] = x; out_pts[i * 3 + 1] = y; out_pts[i * 3 + 2] = z;
  } else {
    // padding rows (incl. one extra tile for pipeline over-read): never win argmin
    memA[i] = make_float4(0.0f, 0.0f, 0.0f, 1e30f);
  }
}

// ---- 2) Bulk copy mem_desc -> out_desc (float4 wide; N*D divisible by 4).
__global__ void k_copy4(const float4* __restrict__ src, float4* __restrict__ dst, int n4) {
  int i = blockIdx.x * blockDim.x + threadIdx.x;
  if (i < n4) dst[i] = src[i];
}

// ---- 3) pix2world: ptsA[p] = (x,y,z,1) valid, else (1e6,1e6,1e6,1).
__global__ void k_pix2world(const float* __restrict__ points, const float* __restrict__ depth,
                            const float* __restrict__ pose, const float* __restrict__ Kmat,
                            float4* __restrict__ ptsA, int* __restrict__ validA,
                            int B, int M, int H, int W) {
  int p = blockIdx.x * blockDim.x + threadIdx.x;
  int P = B * M;
  if (p >= P) return;
  int b = p / M;
  float u = points[p * 2 + 0], v = points[p * 2 + 1];
  // bilinear depth on clipped coords
  float uc = fminf(fmaxf(u, 0.0f), (float)W - 1.001f);
  float vc = fminf(fmaxf(v, 0.0f), (float)H - 1.001f);
  float u0 = floorf(uc), v0 = floorf(vc);
  float du = uc - u0, dv = vc - v0;
  int ui = (int)u0, vi = (int)v0;
  const float* dm = depth + (size_t)b * H * W;
  float d00 = dm[vi * W + ui],       d01 = dm[vi * W + ui + 1];
  float d10 = dm[(vi + 1) * W + ui], d11 = dm[(vi + 1) * W + ui + 1];
  float d = d00 * (1 - du) * (1 - dv) + d01 * du * (1 - dv)
          + d10 * (1 - du) * dv       + d11 * du * dv;
  // 3x3 inverse of K (adjugate)
  const float* K = Kmat + (size_t)b * 9;
  float a00=K[0],a01=K[1],a02=K[2],a10=K[3],a11=K[4],a12=K[5],a20=K[6],a21=K[7],a22=K[8];
  float det = a00*(a11*a22-a12*a21) - a01*(a10*a22-a12*a20) + a02*(a10*a21-a11*a20);
  float id = 1.0f / det;
  float i00=(a11*a22-a12*a21)*id, i01=(a02*a21-a01*a22)*id, i02=(a01*a12-a02*a11)*id;
  float i10=(a12*a20-a10*a22)*id, i11=(a00*a22-a02*a20)*id, i12=(a02*a10-a00*a12)*id;
  float i20=(a10*a21-a11*a20)*id, i21=(a01*a20-a00*a21)*id, i22=(a00*a11-a01*a10)*id;
  // cam = Kinv * (u,v,1) * d  (raw u,v per reference)
  float cx = (i00*u + i01*v + i02) * d;
  float cy = (i10*u + i11*v + i12) * d;
  float cz = (i20*u + i21*v + i22) * d;
  const float* T = pose + (size_t)b * 12; // [3,4] row-major
  float wx = T[0]*cx + T[1]*cy + T[2]*cz  + T[3];
  float wy = T[4]*cx + T[5]*cy + T[6]*cz  + T[7];
  float wz = T[8]*cx + T[9]*cy + T[10]*cz + T[11];
  int valid = (d > 0.0f) && isfinite(wx) && isfinite(wy) && isfinite(wz);
  validA[p] = valid;
  ptsA[p] = valid ? make_float4(wx, wy, wz, 1.0f)
                  : make_float4(1e6f, 1e6f, 1e6f, 1.0f);
}

// ---- 4) WMMA scoring + argmin. One block per 16-point tile; 8 waves take
//         contiguous tile chunks.  score(m,p) = |m|^2 - 2 m.p  via
//         V_WMMA_F32_16X16X4_F32.  Fast path per tile: wmma + 7 fmin + 1 cmp;
//         the (rare) running-min update is the only divergent region.
__global__ __launch_bounds__(256) void k_match(
    const float* __restrict__ memA, const float4* __restrict__ ptsA,
    const int* __restrict__ validA, int* __restrict__ idxA,
    int* __restrict__ matchA, int* __restrict__ unmA, int NT) {
  __shared__ float s_sc[128];
  __shared__ int   s_ix[128];
  const int lane = threadIdx.x & 31;
  const int wave = threadIdx.x >> 5;
  const int col  = lane & 15;     // point within tile / mem row within tile
  const int hi   = lane >> 4;     // 0: K=0,1 half, 1: K=2,3 half

  // B fragment (4x16: points), built once per block.
  float4 pb = ptsA[blockIdx.x * 16 + col];
  v2f bf;
  bf[0] = hi ? pb.z : pb.x;
  bf[1] = hi ? 1.0f : pb.y;

  const int chunk = (NT + 7) >> 3;
  const int tBeg = wave * chunk;
  const int tEnd = (tBeg + chunk < NT) ? tBeg + chunk : NT;

  float bestS = 3.4e38f;
  int   bestI = 0;

  // running A pointer: row (t*16 + col), half (hi)  -> +64 floats per tile
  const float* aptr = memA + ((size_t)tBeg * 16 + col) * 4 + (hi << 1);
  v2f a;
  if (tBeg < tEnd) { a[0] = aptr[0]; a[1] = aptr[1]; }
  for (int t = tBeg; t < tEnd; ++t) {
    // prefetch next tile's fragment (over-read covered by padded memA)
    const float* np = aptr + 64;
    v2f an; an[0] = np[0]; an[1] = np[1];
    v8f c = {0.f, 0.f, 0.f, 0.f, 0.f, 0.f, 0.f, 0.f};
    c = __builtin_amdgcn_wmma_f32_16x16x4_f32(false, a, false, bf,
                                              (short)0, c, false, false);
    float m0 = fminf(c[0], c[1]);
    float m1 = fminf(c[2], c[3]);
    float m2 = fminf(c[4], c[5]);
    float m3 = fminf(c[6], c[7]);
    float tmin = fminf(fminf(m0, m1), fminf(m2, m3));
    if (tmin < bestS) {            // rare after warm-up (~log(NT) times)
      bestS = tmin;
      int sel = 7;
      #pragma unroll
      for (int r = 6; r >= 0; --r) sel = (c[r] == tmin) ? r : sel;  // lowest r
      bestI = (t << 4) + (hi << 3) + sel;
    }
    a = an;
    aptr += 64;
  }
  // merge the two half-rows (same point, mem rows +-8)
  float oS = __shfl_xor(bestS, 16, 32);
  int   oI = __shfl_xor(bestI, 16, 32);
  if (oS < bestS || (oS == bestS && oI < bestI)) { bestS = oS; bestI = oI; }
  if (lane < 16) { s_sc[wave * 16 + lane] = bestS; s_ix[wave * 16 + lane] = bestI; }
  __syncthreads();
  if (threadIdx.x < 16) {
    float bs = s_sc[threadIdx.x]; int bi = s_ix[threadIdx.x];
    #pragma unroll
    for (int w = 1; w < 8; ++w) {
      float v = s_sc[w * 16 + threadIdx.x]; int ix = s_ix[w * 16 + threadIdx.x];
      if (v < bs || (v == bs && ix < bi)) { bs = v; bi = ix; }
    }
    int p = blockIdx.x * 16 + threadIdx.x;
    float4 pp = ptsA[p];
    float d2 = bs + pp.x * pp.x + pp.y * pp.y + pp.z * pp.z;
    int v = validA[p];
    int mt = (d2 < THR2_F) && v;
    idxA[p] = bi; matchA[p] = mt; unmA[p] = (!mt) && v;
  }
}

// ---- 5) Single-block scan over P: slots for unmatched, widx, n_match, n_valid.
__global__ __launch_bounds__(1024) void k_scan(
    const int* __restrict__ unmA, const int* __restrict__ matchA,
    const int* __restrict__ validA, const int* __restrict__ idxA,
    const int* __restrict__ nextPtr, int* __restrict__ widxA,
    int* __restrict__ nvalidOut, float* __restrict__ nmatchOut, int P, int N) {
  __shared__ int s[1024], sm[1024], sv[1024];
  int t = threadIdx.x;
  int base = t * 4;
  int u[4]; int lu = 0, lm = 0, lv = 0;
  #pragma unroll
  for (int j = 0; j < 4; ++j) {
    int p = base + j;
    int uu = (p < P) ? unmA[p] : 0;
    u[j] = uu; lu += uu;
    if (p < P) { lm += matchA[p]; lv += validA[p]; }
  }
  s[t] = lu; sm[t] = lm; sv[t] = lv;
  __syncthreads();
  for (int off = 1; off < 1024; off <<= 1) {
    int v = (t >= off) ? s[t - off] : 0;
    __syncthreads();
    s[t] += v;
    __syncthreads();
  }
  int run = s[t] - lu;       // exclusive prefix of this thread's chunk
  int np = *nextPtr;
  #pragma unroll
  for (int j = 0; j < 4; ++j) {
    int p = base + j;
    if (p < P) {
      run += u[j];
      widxA[p] = u[j] ? ((np + run - 1) % N) : idxA[p];
    }
  }
  __syncthreads();
  for (int o = 512; o > 0; o >>= 1) {
    if (t < o) { sm[t] += sm[t + o]; sv[t] += sv[t + o]; }
    __syncthreads();
  }
  if (t == 0) { *nmatchOut = (float)sm[0]; *nvalidOut = sv[0]; }
}

// ---- 6) Deterministic scatter ownership: highest p with same widx wins.
__global__ void k_owner(const int* __restrict__ validA, const int* __restrict__ widxA,
                        int* __restrict__ ownerA, int P) {
  int p = blockIdx.x * blockDim.x + threadIdx.x;
  if (p >= P) return;
  int ok = validA[p];   // writers are exactly the valid points (match | unm)
  if (ok) {
    int w = widxA[p];
    for (int q = p + 1; q < P; ++q)
      if (validA[q] && widxA[q] == w) { ok = 0; break; }
  }
  ownerA[p] = ok;
}

// ---- 7) Descriptor update + per-point cosine. One wave32 per point.
__global__ __launch_bounds__(256) void k_update(
    const float* __restrict__ desc, const float* __restrict__ mem_desc,
    const float4* __restrict__ ptsA, const int* __restrict__ matchA,
    const int* __restrict__ validA, const int* __restrict__ idxA,
    const int* __restrict__ widxA, const int* __restrict__ ownerA,
    float* __restrict__ out_pts, float* __restrict__ out_desc,
    float* __restrict__ cosArr, int P, int D) {
  int wave = threadIdx.x >> 5;
  int lane = threadIdx.x & 31;
  int p = blockIdx.x * 8 + wave;
  if (p >= P) return;
  const float* dp = desc + (size_t)p * D;
  float d[8];
  #pragma unroll
  for (int j = 0; j < 8; ++j) d[j] = dp[lane + 32 * j];
  float nd2 = 0.f;
  #pragma unroll
  for (int j = 0; j < 8; ++j) nd2 += d[j] * d[j];
  float mxd = fmaxf(sqrtf(wred_sum(nd2)), EPS_F);
  int mt = matchA[p], vl = validA[p], ow = ownerA[p];
  float o[8];
  float cosv;
  if (mt) {
    const float* op = mem_desc + (size_t)idxA[p] * D;
    float no2 = 0.f;
    #pragma unroll
    for (int j = 0; j < 8; ++j) { o[j] = op[lane + 32 * j]; no2 += o[j] * o[j]; }
    float mxo = fmaxf(sqrtf(wred_sum(no2)), EPS_F);
    float acc = 0.f;
    #pragma unroll
    for (int j = 0; j < 8; ++j) acc += (d[j] / mxd) * (o[j] / mxo);
    cosv = wred_sum(acc);
  } else {
    float acc = 0.f;
    #pragma unroll
    for (int j = 0; j < 8; ++j) { float a = d[j] / mxd; acc += a * a; }
    cosv = wred_sum(acc);
  }
  if (lane == 0) cosArr[p] = vl ? cosv : 0.0f;
  if (ow) {
    int w = widxA[p];
    float* outp = out_desc + (size_t)w * D;
    if (mt) {
      float uu[8]; float u2 = 0.f;
      #pragma unroll
      for (int j = 0; j < 8; ++j) { uu[j] = EMA_F * o[j] + (1.0f - EMA_F) * d[j]; u2 += uu[j] * uu[j]; }
      float un = sqrtf(wred_sum(u2)) + EPS_F;
      #pragma unroll
      for (int j = 0; j < 8; ++j) outp[lane + 32 * j] = uu[j] / un;
    } else {
      #pragma unroll
      for (int j = 0; j < 8; ++j) outp[lane + 32 * j] = d[j];
      if (lane < 3) {
        float4 pp = ptsA[p];
        out_pts[w * 3 + lane] = (lane == 0) ? pp.x : (lane == 1) ? pp.y : pp.z;
      }
    }
  }
}

// ---- 8) Fixed-order loss reduction (bitwise deterministic across replays).
__global__ __launch_bounds__(1024) void k_loss(const float* __restrict__ cosArr,
                                               const int* __restrict__ nvalid,
                                               float* __restrict__ out, int P) {
  __shared__ float s[1024];
  int t = threadIdx.x;
  float acc = 0.f;
  for (int j = t; j < P; j += 1024) acc += cosArr[j];
  s[t] = acc;
  __syncthreads();
  for (int o = 512; o > 0; o >>= 1) {
    if (t < o) s[t] += s[t + o];
    __syncthreads();
  }
  if (t == 0) {
    int nv = *nvalid; if (nv < 1) nv = 1;
    out[0] = 1.0f - s[0] / (float)nv;
  }
}

extern "C" void kernel_launch(void* const* d_in, const int* in_sizes, int n_in,
                              void* d_out, int out_size, void* d_ws, size_t ws_size,
                              hipStream_t stream) {
  const float* points   = (const float*)d_in[0];
  const float* depth    = (const float*)d_in[1];
  const float* pose     = (const float*)d_in[2];
  const float* Kmat     = (const float*)d_in[3];
  const float* desc     = (const float*)d_in[4];
  const float* mem_pts  = (const float*)d_in[5];
  const float* mem_desc = (const float*)d_in[6];
  const int*   next_ptr = (const int*)d_in[7];

  const int B = in_sizes[2] / 12;          // pose [B,3,4]
  const int M = in_sizes[0] / (2 * B);     // points [B,M,2]
  const int D = in_sizes[4] / (B * M);     // descriptors [B,M,D]
  const int N = in_sizes[5] / 3;           // mem_pts [N,3]
  const int W = 640;
  const int H = (in_sizes[1] / B) / W;     // depth [B,1,H,W]
  const int P = B * M;
  const int Npad = ((N + 15) / 16) * 16;
  const int NT = Npad / 16;
  const int NpadAll = Npad + 16;           // +1 tile: pipeline over-read safety

  // output layout
  float* out_loss   = (float*)d_out;
  float* out_nmatch = out_loss + 1;
  float* out_p      = out_loss + 2;
  float* out_d      = out_p + (size_t)N * 3;

  // workspace layout (all offsets 256B-aligned; total < 1.1 MB)
  char* ws = (char*)d_ws;
  size_t off = 0;
  auto take = [&](size_t bytes) { void* r = ws + off; off = (off + bytes + 255) & ~(size_t)255; return r; };
  float4* memA   = (float4*)take((size_t)NpadAll * 16);
  float4* ptsA   = (float4*)take((size_t)P * 16);
  int* validA    = (int*)take((size_t)P * 4);
  int* matchA    = (int*)take((size_t)P * 4);
  int* idxA      = (int*)take((size_t)P * 4);
  int* widxA     = (int*)take((size_t)P * 4);
  int* unmA      = (int*)take((size_t)P * 4);
  int* ownerA    = (int*)take((size_t)P * 4);
  float* cosArr  = (float*)take((size_t)P * 4);
  int* nvalidS   = (int*)take(256);
  (void)ws_size;

  // 1) stage WMMA A operand + copy mem_pts to output
  k_prep<<<(NpadAll + 255) / 256, 256, 0, stream>>>(mem_pts, memA, out_p, N, NpadAll);
  // 2) copy mem_desc to output (float4 wide)
  {
    int n4 = (N * D) / 4;
    k_copy4<<<(n4 + 255) / 256, 256, 0, stream>>>((const float4*)mem_desc, (float4*)out_d, n4);
  }
  // 3) back-project points
  k_pix2world<<<(P + 255) / 256, 256, 0, stream>>>(points, depth, pose, Kmat,
                                                   ptsA, validA, B, M, H, W);
  // 4) WMMA distance scoring + argmin over memory
  k_match<<<P / 16, 256, 0, stream>>>((const float*)memA, ptsA, validA,
                                      idxA, matchA, unmA, NT);
  // 5) slot scan + counts
  k_scan<<<1, 1024, 0, stream>>>(unmA, matchA, validA, idxA, next_ptr,
                                 widxA, nvalidS, out_nmatch, P, N);
  // 6) deterministic duplicate resolution
  k_owner<<<(P + 255) / 256, 256, 0, stream>>>(validA, widxA, ownerA, P);
  // 7) descriptor EMA/normalize scatter + per-point cosine
  k_update<<<(P + 7) / 8, 256, 0, stream>>>(desc, mem_desc, ptsA, matchA, validA,
                                            idxA, widxA, ownerA, out_p, out_d,
                                            cosArr, P, D);
  // 8) loss
  k_loss<<<1, 1024, 0, stream>>>(cosArr, nvalidS, out_loss, P);
}